// SpikeFP64Divider_29875792511514
// MI455X (gfx1250) — compile-verified
//
#include <hip/hip_runtime.h>
#include <stdint.h>

// ---------------------------------------------------------------------------
// Gate-exact FP64 soft-divider on bit-vector rows (131072 x 64 float 0/1).
// Bandwidth-bound: 96 MB total traffic -> ~4.1 us floor at 23.3 TB/s.
// Strategy: async global->LDS staging (coalesced 16B chunks, ASYNCcnt),
// per-thread integer emulation of the reference's exact gate semantics
// (including its dead rounding-carry and negative-exponent->inf quirks),
// direct float4 stores of the unpacked result bits.
// ---------------------------------------------------------------------------

#define ROWS_PER_BLOCK 64
#define THREADS 64
#define ROW_PITCH 68  // 64 data floats + 4 pad floats (272B row, 16B aligned)

#if defined(__HIP_DEVICE_COMPILE__) && __has_builtin(__builtin_amdgcn_global_load_async_to_lds_b128)
#define USE_ASYNC_LDS 1
#else
#define USE_ASYNC_LDS 0
#endif

typedef int v4i __attribute__((vector_size(4 * sizeof(int))));

__device__ __forceinline__ void stage_b128(const float* __restrict__ g, float* l) {
#if USE_ASYNC_LDS
  // gfx1250 GLOBAL_LOAD_ASYNC_TO_LDS_B128: per-lane 16B global -> LDS, ASYNCcnt
  __builtin_amdgcn_global_load_async_to_lds_b128(
      (__attribute__((address_space(1))) v4i*)g,
      (__attribute__((address_space(3))) v4i*)l,
      /*offset=*/0, /*cpol=*/0);
#else
  *(float4*)l = *(const float4*)g;
#endif
}

__device__ __forceinline__ uint64_t pack_row(const float* lds_row) {
  uint64_t v = 0;  // bit (63-i) = element i  (MSB-first packing)
#pragma unroll
  for (int k = 0; k < 16; ++k) {
    const uint4 u = *(const uint4*)(lds_row + k * 4);  // ds_load_b128
    uint64_t nib = ((uint64_t)(u.x != 0u) << 3) | ((uint64_t)(u.y != 0u) << 2) |
                   ((uint64_t)(u.z != 0u) << 1) | ((uint64_t)(u.w != 0u));
    v = (v << 4) | nib;
  }
  return v;
}

__global__ __launch_bounds__(THREADS) void fpdiv_bits_kernel(
    const float* __restrict__ A, const float* __restrict__ B,
    float* __restrict__ out, int nrows) {
  __shared__ float lA[ROWS_PER_BLOCK * ROW_PITCH];
  __shared__ float lB[ROWS_PER_BLOCK * ROW_PITCH];

  const int tid  = threadIdx.x;
  const int row0 = blockIdx.x * ROWS_PER_BLOCK;

  // ---- Stage 64 rows of A and B into LDS; consecutive lanes fetch
  // consecutive 16B chunks -> 512B contiguous per wave32 instruction. ----
#pragma unroll
  for (int it = 0; it < (ROWS_PER_BLOCK * 16) / THREADS; ++it) {
    const int c = it * THREADS + tid;    // chunk id in block: 0..1023
    const int r = c >> 4;                // row within block
    const int k = c & 15;                // 16B chunk within row
    if (row0 + r < nrows) {
      const size_t goff = (size_t)(row0 + r) * 64 + (size_t)k * 4;
      stage_b128(A + goff, &lA[r * ROW_PITCH + k * 4]);
      stage_b128(B + goff, &lB[r * ROW_PITCH + k * 4]);
    }
  }
#if USE_ASYNC_LDS
#if __has_builtin(__builtin_amdgcn_s_wait_asynccnt)
  __builtin_amdgcn_s_wait_asynccnt(0);
#else
  asm volatile("s_wait_asynccnt 0" ::: "memory");
#endif
#endif
  __syncthreads();

  const int row = row0 + tid;
  if (row >= nrows) return;

  const uint64_t va = pack_row(&lA[tid * ROW_PITCH]);
  const uint64_t vb = pack_row(&lB[tid * ROW_PITCH]);

  const uint64_t M52 = 0xFFFFFFFFFFFFFull;
  const uint32_t sa = (uint32_t)(va >> 63);
  const uint32_t sb = (uint32_t)(vb >> 63);
  const uint32_t ea = (uint32_t)(va >> 52) & 0x7FFu;
  const uint32_t eb = (uint32_t)(vb >> 52) & 0x7FFu;
  const uint64_t ma = va & M52;
  const uint64_t mb = vb & M52;

  const bool a_zero = (ea == 0u) && (ma == 0);
  const bool b_zero = (eb == 0u) && (mb == 0);
  const bool a_inf  = (ea == 0x7FFu) && (ma == 0);
  const bool b_inf  = (eb == 0x7FFu) && (mb == 0);
  const bool a_nan  = (ea == 0x7FFu) && (ma != 0);
  const bool b_nan  = (eb == 0x7FFu) && (mb != 0);

  const uint32_t s_out = sa ^ sb;
  const bool r_nan  = a_nan || b_nan || (a_zero && b_zero) || (a_inf && b_inf);
  const bool r_inf  = (!a_zero && b_zero) || (a_inf && !b_inf);
  const bool r_zero = (a_zero && !b_zero) || (!a_inf && b_inf);

  // 13-bit mod-8192 exponent math (reference drops borrows/carries).
  const uint32_t exp13 = (ea - eb + 1023u) & 0x1FFFu;

  // 55-bit restoring division: A_55 = 0,1,m_a,0 ; D_55 = 0,1,m_b,0 (MSB-first)
  const uint64_t Dq = (1ull << 53) | (mb << 1);
  uint64_t R = (1ull << 53) | (ma << 1);
  uint64_t Q = 0;  // 57 quotient bits, first bit ends at position 56
#pragma unroll
  for (int i = 0; i < 57; ++i) {
    const uint64_t t = R - Dq;
    const uint64_t ge = (R >= Dq) ? 1ull : 0ull;
    Q = (Q << 1) | ge;
    R = ge ? t : R;
    if (i < 56) R <<= 1;  // MSB provably 0 here (R < D < 2^54 after select)
  }
  const bool rem_nz = (R != 0);

  const uint32_t q0 = (uint32_t)(Q >> 56) & 1u;
  // normal path (q0==1): mant=Q[1:53], round=Q[53], sticky=Q[54:57]|rem
  // shifted path (q0==0): mant=Q[2:54], round=Q[54], sticky=Q[55:57]|rem, exp-1
  const uint64_t mant  = q0 ? ((Q >> 4) & M52) : ((Q >> 3) & M52);
  const uint32_t rnd   = q0 ? ((uint32_t)(Q >> 3) & 1u) : ((uint32_t)(Q >> 2) & 1u);
  const bool sticky    = q0 ? (((Q & 7ull) != 0) || rem_nz)
                            : (((Q & 3ull) != 0) || rem_nz);
  const uint32_t expadj = q0 ? exp13 : ((exp13 - 1u) & 0x1FFFu);

  const uint32_t lsb = (uint32_t)mant & 1u;
  const uint32_t round_up = rnd & ((sticky || lsb) ? 1u : 0u);
  // Reference quirk: ripple_add's carry-out over 53 bits is always 0, so a
  // rounding overflow wraps mantissa to 0 WITHOUT bumping the exponent.
  const uint64_t mant_f = (mant + round_up) & M52;
  const uint32_t exp_f13 = expadj;

  const bool ovf = (exp_f13 & 0x1800u) != 0;                       // bits 11|12
  const bool unf = (((exp_f13 >> 12) & 1u) != 0) || (exp_f13 == 0);

  uint64_t res = ((uint64_t)s_out << 63) |
                 ((uint64_t)(exp_f13 & 0x7FFu) << 52) | mant_f;
  const uint64_t infv  = ((uint64_t)s_out << 63) | (0x7FFull << 52);
  const uint64_t nanv  = infv | (1ull << 51);
  const uint64_t zerov = ((uint64_t)s_out << 63);
  if (r_nan)       res = nanv;
  else if (r_inf)  res = infv;
  else if (r_zero) res = zerov;
  else if (ovf)    res = infv;   // note: negative exponents land here (bit12)
  else if (unf)    res = zerov;

  // Unpack MSB-first to 64 output floats (0.0 / 1.0).
  float* orow = out + (size_t)row * 64;
#pragma unroll
  for (int k = 0; k < 16; ++k) {
    float4 f;
    f.x = ((res >> (63 - (4 * k + 0))) & 1ull) ? 1.0f : 0.0f;
    f.y = ((res >> (63 - (4 * k + 1))) & 1ull) ? 1.0f : 0.0f;
    f.z = ((res >> (63 - (4 * k + 2))) & 1ull) ? 1.0f : 0.0f;
    f.w = ((res >> (63 - (4 * k + 3))) & 1ull) ? 1.0f : 0.0f;
    *(float4*)(orow + 4 * k) = f;
  }
}

extern "C" void kernel_launch(void* const* d_in, const int* in_sizes, int n_in,
                              void* d_out, int out_size, void* d_ws, size_t ws_size,
                              hipStream_t stream) {
  (void)n_in; (void)out_size; (void)d_ws; (void)ws_size;
  const float* A = (const float*)d_in[0];
  const float* B = (const float*)d_in[1];
  float* out = (float*)d_out;
  const int nrows = in_sizes[0] / 64;  // 131072
  const int blocks = (nrows + ROWS_PER_BLOCK - 1) / ROWS_PER_BLOCK;
  hipLaunchKernelGGL(fpdiv_bits_kernel, dim3(blocks), dim3(THREADS), 0, stream,
                     A, B, out, nrows);
}